// GLNonlinSlow_56427280335279
// MI455X (gfx1250) — compile-verified
//
#include <hip/hip_runtime.h>

// Shapes fixed by the reference: uvs [16, 16, 512, 256] fp32.
// BC = 16*16 = 256 flattened (b, channel) pairs; channel c pairs with c^8.
#define M_DIM 512
#define D_DIM 256

typedef __attribute__((ext_vector_type(2))) float v2f;
typedef __attribute__((ext_vector_type(8))) float v8f;

// ---------------------------------------------------------------------------
// Pass 1: sums[bc][d] = sum over m of uvs[bc][m][d].
// Grid: 1024 blocks = 256 (bc) x 4 (quarter of d). Block: 256 threads.
// Thread layout: cg = tid&15 -> 4-col group within the 64-col quarter,
//                ro = tid>>4 -> m phase (0..15). float4 loads, LDS tree reduce.
// ---------------------------------------------------------------------------
__global__ void __launch_bounds__(256)
colsum_kernel(const float* __restrict__ uvs, float* __restrict__ sums) {
    const int blk = blockIdx.x;
    const int bc  = blk >> 2;          // 0..255
    const int q   = blk & 3;           // d quarter
    const int tid = threadIdx.x;
    const int cg  = tid & 15;          // col group (4 floats)
    const int ro  = tid >> 4;          // 0..15 row phase

    const float4* base = (const float4*)(uvs + (size_t)bc * M_DIM * D_DIM);
    const int c4 = q * 16 + cg;        // float4 column index, 0..63

    float4 acc = make_float4(0.f, 0.f, 0.f, 0.f);
    #pragma unroll 4
    for (int m = ro; m < M_DIM; m += 16) {
        float4 v = base[(size_t)m * (D_DIM / 4) + c4];
        acc.x += v.x; acc.y += v.y; acc.z += v.z; acc.w += v.w;
    }

    __shared__ float4 red[256];
    red[tid] = acc;
    __syncthreads();

    if (ro == 0) {
        float4 t = red[cg];
        #pragma unroll
        for (int r = 1; r < 16; ++r) {
            float4 v = red[cg + 16 * r];
            t.x += v.x; t.y += v.y; t.z += v.z; t.w += v.w;
        }
        ((float4*)(sums + (size_t)bc * D_DIM))[c4] = t;
    }
}

// ---------------------------------------------------------------------------
// Pass 2: per row m of channel bc: g = (dot(row, sums[bc^8]) > 0) ? 1 : 0,
//         out row = g * row.
// Grid: 1024 blocks = 256 (bc) x 4 (tile of 128 rows). Block: 256 threads.
// Each of the 8 waves computes 16 row-dots with V_WMMA_F32_16X16X4_F32:
//   A = 16 rows x K=4 slice of the tile, B = partner-sum slice replicated
//   into all 16 columns, C accumulates over 64 steps (K=256).
// A layout (ISA 7.12.2, 32-bit A 16x4): lanes 0-15 hold {K0,K1} of row M=lane,
//   lanes 16-31 hold {K2,K3} of row M=lane-16. B (4x16) mirrors with K in the
//   same slot/half positions; since all 16 B columns are identical, the
//   lane->N mapping is irrelevant.
// D layout: acc[r], lanes 0-15 -> row r; lanes 16-31 -> row 8+r. So lanes 0
//   and 16 each hold 8 dots -> gates to LDS -> coalesced float4 scale pass.
// ---------------------------------------------------------------------------
__global__ void __launch_bounds__(256)
gate_scale_kernel(const float* __restrict__ uvs, const float* __restrict__ sums,
                  float* __restrict__ out) {
    const int blk  = blockIdx.x;
    const int bc   = blk >> 2;         // 0..255
    const int mt   = blk & 3;          // 128-row tile
    const int tid  = threadIdx.x;
    const int wave = tid >> 5;         // 0..7
    const int lane = tid & 31;

    const int m0 = mt * 128 + wave * 16;        // this wave's 16-row tile
    const float* S = sums + (size_t)(bc ^ 8) * D_DIM;   // partner column-sum

    const int half   = lane >> 4;      // 0: K pair {0,1}; 1: K pair {2,3}
    const int l15    = lane & 15;      // row within tile for A
    const int colOff = half * 2;
    const float* arow = uvs + ((size_t)bc * M_DIM + (m0 + l15)) * D_DIM;

    v8f acc = (v8f)0.0f;
    #pragma unroll 8
    for (int j = 0; j < 64; ++j) {
        const int c = 4 * j + colOff;
        v2f a = *(const v2f*)(arow + c);
        v2f b = *(const v2f*)(S + c);
        acc = __builtin_amdgcn_wmma_f32_16x16x4_f32(
            /*neg_a=*/false, a, /*neg_b=*/false, b,
            /*c_mod=*/(short)0, acc, /*reuse_a=*/false, /*reuse_b=*/false);
    }

    __shared__ float gates[128];
    if (l15 == 0) {  // lane 0 -> rows 0..7 of wave tile, lane 16 -> rows 8..15
        const int rbase = wave * 16 + half * 8;
        #pragma unroll
        for (int r = 0; r < 8; ++r)
            gates[rbase + r] = acc[r] > 0.0f ? 1.0f : 0.0f;
    }
    __syncthreads();

    // Scale the 128x256 tile (hot in WGP$/L2 from the WMMA reads).
    const size_t tbase = ((size_t)bc * M_DIM + mt * 128) * D_DIM;
    const float4* src = (const float4*)(uvs + tbase);
    float4*       dst = (float4*)(out + tbase);
    #pragma unroll 4
    for (int i = tid; i < 128 * (D_DIM / 4); i += 256) {
        const float g = gates[i >> 6];   // 64 float4 per row
        float4 v = src[i];
        v.x *= g; v.y *= g; v.z *= g; v.w *= g;
        dst[i] = v;
    }
}

// ---------------------------------------------------------------------------
extern "C" void kernel_launch(void* const* d_in, const int* in_sizes, int n_in,
                              void* d_out, int out_size, void* d_ws, size_t ws_size,
                              hipStream_t stream) {
    (void)in_sizes; (void)n_in; (void)out_size; (void)ws_size;
    const float* uvs  = (const float*)d_in[0];
    float*       out  = (float*)d_out;
    float*       sums = (float*)d_ws;   // 256 * 256 floats = 256 KB scratch

    colsum_kernel<<<dim3(1024), dim3(256), 0, stream>>>(uvs, sums);
    gate_scale_kernel<<<dim3(1024), dim3(256), 0, stream>>>(uvs, sums, out);
}